// CrossModalAttention2D_87565793231371
// MI455X (gfx1250) — compile-verified
//
#include <hip/hip_runtime.h>
#include <hip/hip_bf16.h>

// CrossModalAttention2D collapses: KV length == 1 -> softmax == 1 -> attention
// output == v broadcast. Only the text path + two projections survive, plus a
// streaming residual add over the 134MB visual tensor (memory-bound pass).

#define C_DIM 1024
#define B_DIM 8
#define NT_DIM 8
#define HW_DIM 4096

typedef float v2f __attribute__((ext_vector_type(2)));
typedef float v4f __attribute__((ext_vector_type(4)));
typedef float v8f __attribute__((ext_vector_type(8)));

// ---------------------------------------------------------------------------
// Kernel 1: t[b][c] = mean over NT of text[b, nt, c], padded to 16 rows (zeros)
// ---------------------------------------------------------------------------
__global__ void text_mean_kernel(const float* __restrict__ text,
                                 float* __restrict__ t_pad) {
    int idx = blockIdx.x * blockDim.x + threadIdx.x;   // 0 .. 16*1024-1
    if (idx >= 16 * C_DIM) return;
    int b = idx >> 10;          // padded batch row
    int c = idx & (C_DIM - 1);
    float s = 0.0f;
    if (b < B_DIM) {
        const float* p = text + b * (NT_DIM * C_DIM) + c;
#pragma unroll
        for (int nt = 0; nt < NT_DIM; ++nt) s += p[nt * C_DIM];
        s *= (1.0f / NT_DIM);
    }
    t_pad[idx] = s;             // rows 8..15 zeroed (ws is poisoned 0xAA)
}

// ---------------------------------------------------------------------------
// Kernel 2: O(16,1024) = A(16,1024) @ W(1024,1024)^T + bias, fp32 WMMA 16x16x4.
// One wave per 16-wide N tile; 64 waves total (16 blocks x 4 waves).
// ---------------------------------------------------------------------------
__global__ void gemv16_wmma_kernel(const float* __restrict__ A,
                                   const float* __restrict__ W,
                                   const float* __restrict__ bias,
                                   float* __restrict__ O) {
    const int K = C_DIM;
    int lane  = threadIdx.x & 31;
    int wave  = threadIdx.x >> 5;
    int ntile = blockIdx.x * (blockDim.x >> 5) + wave;  // 0..63
    int nbase = ntile * 16;

    int mrow  = lane & 15;            // A: M index for this lane
    int khalf = (lane >> 4) << 1;     // A: lanes 16-31 hold K+2,K+3
    int ncol  = lane & 15;            // B/D: N index for this lane
    int krow  = lane >> 4;            // B: VGPR0 holds K row {0,1}

    const float* Arow = A + mrow * K + khalf;
    const float* Wrow = W + (nbase + ncol) * K + krow;

    v8f acc = {};
    for (int k0 = 0; k0 < K; k0 += 4) {
        v2f a, b;
        a.x = Arow[k0];
        a.y = Arow[k0 + 1];
        b.x = Wrow[k0];        // B[krow   ][n] = W[nbase+n][k0+krow]
        b.y = Wrow[k0 + 2];    // B[krow+2 ][n]
        acc = __builtin_amdgcn_wmma_f32_16x16x4_f32(
            /*neg_a=*/false, a, /*neg_b=*/false, b,
            /*c_mod=*/(short)0, acc, /*reuse_a=*/false, /*reuse_b=*/false);
    }

    float bv  = bias[nbase + ncol];
    int   mhi = (lane >> 4) * 8;      // D: VGPR r -> M = r + 8*(lane>>4)
#pragma unroll
    for (int r = 0; r < 8; ++r) {
        O[(mhi + r) * C_DIM + nbase + ncol] = acc[r] + bv;
    }
}

// ---------------------------------------------------------------------------
// Kernel 3: out[b,c,hw] = visual[b,c,hw] + gamma * o2[b,c]
// Streaming, 128-bit vectorized, non-temporal (268 MB streamed once).
// ---------------------------------------------------------------------------
__global__ void residual_add_kernel(const v4f* __restrict__ visual,
                                    const float* __restrict__ o2_pad,
                                    const float* __restrict__ gamma,
                                    v4f* __restrict__ out) {
    int idx4 = blockIdx.x * blockDim.x + threadIdx.x;  // one v4f each
    // 1024 v4f per (b,c) row of HW=4096 floats
    int row = idx4 >> 10;               // global row = b*C + c
    int b   = row >> 10;
    int c   = row & (C_DIM - 1);
    float add = gamma[0] * o2_pad[b * C_DIM + c];

    v4f v = __builtin_nontemporal_load(&visual[idx4]);
    v.x += add; v.y += add; v.z += add; v.w += add;
    __builtin_nontemporal_store(v, &out[idx4]);
}

// ---------------------------------------------------------------------------
extern "C" void kernel_launch(void* const* d_in, const int* in_sizes, int n_in,
                              void* d_out, int out_size, void* d_ws, size_t ws_size,
                              hipStream_t stream) {
    const float* visual    = (const float*)d_in[0];
    const float* text      = (const float*)d_in[1];
    const float* in_proj_w = (const float*)d_in[2];
    const float* in_proj_b = (const float*)d_in[3];
    const float* out_w     = (const float*)d_in[4];
    const float* out_b     = (const float*)d_in[5];
    // d_in[6], d_in[7]: ln_w / ln_b -- dead (softmax over size-1 axis)
    const float* proj_w    = (const float*)d_in[8];
    const float* proj_b    = (const float*)d_in[9];
    const float* gamma     = (const float*)d_in[10];
    float*       out       = (float*)d_out;

    float* ws = (float*)d_ws;
    float* t_pad  = ws;                 // [16][1024]
    float* v_pad  = ws + 16 * C_DIM;    // [16][1024]
    float* o1_pad = ws + 32 * C_DIM;    // [16][1024]
    float* o2_pad = ws + 48 * C_DIM;    // [16][1024]

    // 1) mean over text tokens (padded to 16 rows)
    text_mean_kernel<<<(16 * C_DIM) / 256, 256, 0, stream>>>(text, t_pad);

    // 2) chained fp32 WMMA GEMVs: v = t@Wv^T+bv ; o1 = v@out_w^T+out_b ; o2 = o1@proj_w^T+proj_b
    const float* Wv = in_proj_w + 2 * C_DIM * C_DIM;  // rows 2C..3C
    const float* bv = in_proj_b + 2 * C_DIM;
    gemv16_wmma_kernel<<<16, 128, 0, stream>>>(t_pad,  Wv,     bv,     v_pad);
    gemv16_wmma_kernel<<<16, 128, 0, stream>>>(v_pad,  out_w,  out_b,  o1_pad);
    gemv16_wmma_kernel<<<16, 128, 0, stream>>>(o1_pad, proj_w, proj_b, o2_pad);

    // 3) streaming residual add: 8*1024*4096 floats = 8388608 v4f's
    const int n4 = (B_DIM * C_DIM * HW_DIM) / 4;
    residual_add_kernel<<<n4 / 256, 256, 0, stream>>>(
        (const v4f*)visual, o2_pad, gamma, (v4f*)out);
}